// GreedyRouter_27273042330075
// MI455X (gfx1250) — compile-verified
//
#include <hip/hip_runtime.h>
#include <cstdint>
#include <cstddef>

#define T_TOKENS  262144
#define N_EXP     256
#define TOP_K     8
#define WAVE      32
#define BLOCK     256
#define WPB       (BLOCK / WAVE)   // 8 waves per block
#define NBLOCKS   1024             // 8192 waves -> 32 tokens/wave

// ---- CDNA5 async global<->LDS path (guarded; fallback = direct b128) ----
#if defined(__has_builtin)
#if __has_builtin(__builtin_amdgcn_global_load_async_to_lds_b128) && \
    __has_builtin(__builtin_amdgcn_global_store_async_from_lds_b128)
#define HAVE_ASYNC_LDS 1
#endif
#endif

#if defined(__has_builtin)
#if __has_builtin(__builtin_amdgcn_s_wait_asynccnt)
#define WAIT_ASYNC(n) __builtin_amdgcn_s_wait_asynccnt(n)
#else
#define WAIT_ASYNC(n) asm volatile("s_wait_asynccnt %0" ::"i"(n) : "memory")
#endif
#else
#define WAIT_ASYNC(n) asm volatile("s_wait_asynccnt %0" ::"i"(n) : "memory")
#endif

// 16-byte int vector in global (AS1) and LDS (AS3) address spaces,
// matching the async builtins' parameter types.
typedef int v4i __attribute__((__vector_size__(16)));
typedef __attribute__((address_space(1))) v4i GV4;
typedef __attribute__((address_space(3))) v4i LV4;

__global__ __launch_bounds__(BLOCK) void moe_router_kernel(
    const float* __restrict__ logits,
    float* __restrict__ out,
    unsigned* __restrict__ ws_hist)
{
    __shared__ float    lbuf[WPB][2][N_EXP];  // 16 KB: double-buffered row staging
    __shared__ unsigned hist[N_EXP];          // per-block expert counts

    const int tid = threadIdx.x;
    hist[tid] = 0u;                            // BLOCK == N_EXP == 256
    __syncthreads();

    const int    lane   = tid & (WAVE - 1);
    const int    wv     = tid >> 5;
    const size_t nwaves = (size_t)gridDim.x * WPB;

    float* out_logits = out;
    float* out_rw     = out + (size_t)T_TOKENS * N_EXP;
    float* out_tw     = out_rw + (size_t)T_TOKENS * N_EXP;
    float* out_ti     = out_tw + (size_t)T_TOKENS * TOP_K;

#if defined(HAVE_ASYNC_LDS)
    // DMA one 1KB row (2 x 32 lanes x 16B); INST_OFFSET applies to BOTH the
    // global and the LDS address, so the second half is just offset 512.
    auto issue_load = [&](const float* src, float* dstlds) {
        GV4* g = (GV4*)(uintptr_t)src + lane;
        LV4* l = (LV4*)(uintptr_t)dstlds + lane;
        __builtin_amdgcn_global_load_async_to_lds_b128(g, l, 0,   0);
        __builtin_amdgcn_global_load_async_to_lds_b128(g, l, 512, 0);
    };
    auto issue_copy = [&](float* dstg, const float* srclds) {
        GV4* g = (GV4*)(uintptr_t)dstg + lane;
        LV4* l = (LV4*)(uintptr_t)srclds + lane;
        __builtin_amdgcn_global_store_async_from_lds_b128(g, l, 0,   0);
        __builtin_amdgcn_global_store_async_from_lds_b128(g, l, 512, 0);
    };
#endif

    size_t t  = (size_t)blockIdx.x * WPB + wv;
    int    pb = 0;

#if defined(HAVE_ASYNC_LDS)
    if (t < T_TOKENS) issue_load(logits + t * (size_t)N_EXP, &lbuf[wv][0][0]);
#endif

    for (; t < T_TOKENS; t += nwaves, pb ^= 1) {
        const size_t tn   = t + nwaves;
        float*       dst1 = out_rw + t * (size_t)N_EXP;

        // Warm L2 two rows ahead of the DMA pipeline (global_prefetch_b8).
        if (tn + nwaves < T_TOKENS)
            __builtin_prefetch(logits + (tn + nwaves) * (size_t)N_EXP, 0, 0);

        float x[8];
#if defined(HAVE_ASYNC_LDS)
        {
            float* cur = &lbuf[wv][pb][0];
            // Drain: current row's loads + previous iteration's copy-stores.
            WAIT_ASYNC(0);
            // Logits copy streams LDS->global, never touching VGPRs.
            issue_copy(out_logits + t * (size_t)N_EXP, cur);
            // Prefetch next row into the other buffer; overlaps compute below.
            if (tn < T_TOKENS) issue_load(logits + tn * (size_t)N_EXP, &lbuf[wv][pb ^ 1][0]);

            const float4 a = *(const float4*)(cur + lane * 8);
            const float4 b = *(const float4*)(cur + lane * 8 + 4);
            x[0]=a.x; x[1]=a.y; x[2]=a.z; x[3]=a.w;
            x[4]=b.x; x[5]=b.y; x[6]=b.z; x[7]=b.w;
        }
#else
        {
            const float* src  = logits + t * (size_t)N_EXP;
            float*       dst0 = out_logits + t * (size_t)N_EXP;
            const float4 a = *(const float4*)(src + lane * 8);
            const float4 b = *(const float4*)(src + lane * 8 + 4);
            *(float4*)(dst0 + lane * 8)     = a;
            *(float4*)(dst0 + lane * 8 + 4) = b;
            x[0]=a.x; x[1]=a.y; x[2]=a.z; x[3]=a.w;
            x[4]=b.x; x[5]=b.y; x[6]=b.z; x[7]=b.w;
        }
#endif
        // ---- softmax over 256 experts (wave32 shuffle reductions) ----
        float m = x[0];
#pragma unroll
        for (int j = 1; j < 8; ++j) m = fmaxf(m, x[j]);
#pragma unroll
        for (int off = 16; off >= 1; off >>= 1)
            m = fmaxf(m, __shfl_xor(m, off, WAVE));

        float s = 0.0f;
#pragma unroll
        for (int j = 0; j < 8; ++j) { x[j] = __expf(x[j] - m); s += x[j]; }
#pragma unroll
        for (int off = 16; off >= 1; off >>= 1)
            s += __shfl_xor(s, off, WAVE);

        const float inv = 1.0f / s;
#pragma unroll
        for (int j = 0; j < 8; ++j) x[j] *= inv;

        *(float4*)(dst1 + lane * 8)     = make_float4(x[0], x[1], x[2], x[3]);
        *(float4*)(dst1 + lane * 8 + 4) = make_float4(x[4], x[5], x[6], x[7]);

        // ---- top-8: packed (value_bits << 32) | ~id ; lower id wins ties ----
        unsigned long long key[8];
#pragma unroll
        for (int j = 0; j < 8; ++j)
            key[j] = ((unsigned long long)__float_as_uint(x[j]) << 32) |
                     (unsigned)(~(unsigned)(lane * 8 + j));

        float sumk = 0.0f, twv = 0.0f, tiv = 0.0f;
#pragma unroll
        for (int k = 0; k < TOP_K; ++k) {
            unsigned long long best = key[0];
#pragma unroll
            for (int j = 1; j < 8; ++j) best = (key[j] > best) ? key[j] : best;
#pragma unroll
            for (int off = 16; off >= 1; off >>= 1) {
                unsigned long long o = __shfl_xor(best, off, WAVE);
                best = (o > best) ? o : best;
            }
            const unsigned idu = ~(unsigned)best;                       // expert id
            const float    val = __uint_as_float((unsigned)(best >> 32));
            sumk += val;
            if (lane == k) {
                twv = val;
                tiv = (float)idu;
                atomicAdd(&hist[idu], 1u);                              // LDS atomic
            }
            const bool     mine = ((idu >> 3) == (unsigned)lane);
            const unsigned jrm  = idu & 7u;
#pragma unroll
            for (int q = 0; q < 8; ++q)
                key[q] = (mine && (jrm == (unsigned)q)) ? 0ull : key[q];
        }

        const float nrm = 1.0f / sumk;
        if (lane < TOP_K) {
            out_tw[t * TOP_K + lane] = twv * nrm;
            out_ti[t * TOP_K + lane] = tiv;
        }
    }

    __syncthreads();
    atomicAdd(&ws_hist[tid], hist[tid]);   // one global atomic per expert per block
    // trailing async copy-stores drained by s_endpgm's implicit wait-idle
}

__global__ void zero_ws_kernel(unsigned* __restrict__ ws) {
    ws[threadIdx.x] = 0u;
}

__global__ void finalize_hist_kernel(const unsigned* __restrict__ ws,
                                     float* __restrict__ out_hist) {
    out_hist[threadIdx.x] = (float)ws[threadIdx.x];
}

extern "C" void kernel_launch(void* const* d_in, const int* in_sizes, int n_in,
                              void* d_out, int out_size, void* d_ws, size_t ws_size,
                              hipStream_t stream) {
    const float* logits = (const float*)d_in[0];
    float*       out    = (float*)d_out;
    unsigned*    ws     = (unsigned*)d_ws;

    float* out_hist = out + 2 * (size_t)T_TOKENS * N_EXP + 2 * (size_t)T_TOKENS * TOP_K;

    zero_ws_kernel<<<1, N_EXP, 0, stream>>>(ws);
    moe_router_kernel<<<NBLOCKS, BLOCK, 0, stream>>>(logits, out, ws);
    finalize_hist_kernel<<<1, N_EXP, 0, stream>>>(ws, out_hist);
}